// Decoder_50792283243022
// MI455X (gfx1250) — compile-verified
//
#include <hip/hip_runtime.h>

// ---------------------------------------------------------------------------
// Tacotron-style attention decoder for MI455X (gfx1250), compile-only target.
//   kernel 1: p_annot = inputs @ annot_w^T + annot_b   (big parallel GEMM, WMMA f32)
//   kernel 2: Weff    = loc_lin_w @ conv_w folded      (tiny)
//   kernel 3: persistent single-workgroup decoder, 100 sequential steps,
//             all recurrent state + GEMM scratch in 313KB of LDS,
//             every matmul via V_WMMA_F32_16X16X4_F32 (exact f32).
// ---------------------------------------------------------------------------

typedef __attribute__((ext_vector_type(2))) float v2f;
typedef __attribute__((ext_vector_type(8))) float v8f;

#define B_     32
#define T_     512
#define IN_    256
#define MD_    80
#define TDEC   100
#define H_     256
#define A_     128
#define MELN   560          // MD*R
#define KPRE   560          // MS*MD

// ---- workspace layout (floats) --------------------------------------------
#define WS_PANNOT 0
#define WS_PLOC   (WS_PANNOT + (size_t)B_*T_*A_)        // 2M floats
#define WS_WEFF   (WS_PLOC   + (size_t)B_*T_*A_)        // +2M
#define WS_ATT    (WS_WEFF + 128*64)
#define WS_ATTC   (WS_ATT + B_*T_)
// total ~4.25M floats (~17MB)

// ---- d_out layout (floats): outs | attns | stops ---------------------------
#define OUT_OUTS  0
#define OUT_ATTNS ((size_t)B_*TDEC*MELN)
#define OUT_STOPS (OUT_ATTNS + (size_t)B_*TDEC*T_)

// ---- LDS layout (floats). ld = 258/386/514/770 keep stride%64 == 2 (562%64 == 50) ----
constexpr int L_HA   = 0;                   // attn GRU h   [32][258]
constexpr int L_HD1  = L_HA  + 32*258;      // dec GRU1 h   [32][258]
constexpr int L_HD2  = L_HD1 + 32*258;      // dec GRU2 h   [32][258]
constexpr int L_CTX  = L_HD2 + 32*258;      // context      [32][258]
constexpr int L_RED  = L_CTX + 32*258;      // reductions   [64]
constexpr int L_XBUF = L_RED + 64;          // [32][386] arnn x / decoder din
constexpr int L_GI   = L_XBUF + 32*386;     // [32][770] gates; aliases: memin[32][562],
                                            //   pq[32][130]+v[128], cbuf[32][514], mel[32][562]
constexpr int L_GHB  = L_GI + 32*770;       // [32][258] gh slice / pm1 / deco
constexpr int L_TOT  = L_GHB + 32*258;      // 78336 floats = 313,344 bytes (<320KB WGP LDS)

// ---------------------------------------------------------------------------
// f32 WMMA GEMM: C[m][n] = act( sum_k A[m][k] * W[n][k] + bias[n] )
// A in LDS (lda even), W row-major [N][K] in global (K even), C in LDS.
// M fixed to 32 (two 16-row tiles). One 16x16 output tile per wave.
// ---------------------------------------------------------------------------
__device__ __forceinline__ void gemm_lds(const float* lds_, int aoff, int lda,
                                         const float* __restrict__ W,
                                         const float* __restrict__ bias,
                                         float* ldsw, int coff, int ldc,
                                         int N, int K, int act, int wave, int lane)
{
    const int nt = N >> 4, tiles = 2 * nt;
    const int row = lane & 15, half = lane >> 4;
    for (int t = wave; t < tiles; t += 32) {
        const int m0 = (t & 1) << 4;
        const int n0 = (t >> 1) << 4;
        v8f c = {0.f, 0.f, 0.f, 0.f, 0.f, 0.f, 0.f, 0.f};
        const float* ap = lds_ + aoff + (m0 + row) * lda + 2 * half;
        const float* wp = W + (size_t)(n0 + row) * K + 2 * half;
        for (int k = 0; k < K; k += 4) {
            v2f a = *(const v2f*)(ap + k);            // A[row][k+2h .. +1]
            v2f b = *(const v2f*)(wp + k);            // W[n0+col][k+2h .. +1]
            c = __builtin_amdgcn_wmma_f32_16x16x4_f32(false, a, false, b,
                                                      (short)0, c, false, false);
        }
        const int col = n0 + row;
        const float bv = bias ? bias[col] : 0.f;
#pragma unroll
        for (int i = 0; i < 8; ++i) {
            float v = c[i] + bv;
            if (act == 1)      v = fmaxf(v, 0.f);
            else if (act == 2) v = 1.f / (1.f + expf(-v));
            ldsw[coff + (m0 + i + 8 * half) * ldc + col] = v;
        }
    }
}

// GRU cell, state h in LDS [32][258], x in LDS [32][ldx] (K valid cols).
// gi = x@wih^T+bih computed whole (768 wide); gh computed in 256-wide r/z/n
// slices and fused with gate math to fit LDS.  h updated in place.
__device__ __forceinline__ void gru(float* lds, int hoff, int xoff, int ldx, int K,
                                    const float* wih, const float* whh,
                                    const float* bih, const float* bhh,
                                    int wave, int lane, int tid)
{
    gemm_lds(lds, xoff, ldx, wih, bih, lds, L_GI, 770, 768, K, 0, wave, lane);
    __syncthreads();
    for (int s = 0; s < 2; ++s) {                       // r then z
        gemm_lds(lds, hoff, 258, whh + (size_t)s * 256 * 256, bhh + s * 256,
                 lds, L_GHB, 258, 256, 256, 0, wave, lane);
        __syncthreads();
        for (int idx = tid; idx < 32 * 256; idx += 1024) {
            int b = idx >> 8, j = idx & 255;
            float g = lds[L_GI + b * 770 + s * 256 + j] + lds[L_GHB + b * 258 + j];
            lds[L_GI + b * 770 + s * 256 + j] = 1.f / (1.f + expf(-g));
        }
        __syncthreads();
    }
    gemm_lds(lds, hoff, 258, whh + (size_t)2 * 256 * 256, bhh + 512,
             lds, L_GHB, 258, 256, 256, 0, wave, lane);
    __syncthreads();
    for (int idx = tid; idx < 32 * 256; idx += 1024) {  // n + state update
        int b = idx >> 8, j = idx & 255;
        float r = lds[L_GI + b * 770 + j];
        float z = lds[L_GI + b * 770 + 256 + j];
        float nn = tanhf(lds[L_GI + b * 770 + 512 + j] + r * lds[L_GHB + b * 258 + j]);
        float h = lds[hoff + b * 258 + j];
        lds[hoff + b * 258 + j] = (1.f - z) * nn + z * h;
    }
    __syncthreads();
}

// im2col sample for folded location conv: K-dim = [31 taps of attn | 31 of attn_cum]
__device__ __forceinline__ float im2col(const float* att, const float* attc,
                                        int b, int t, int k)
{
    if (k >= 62) return 0.f;
    const int c  = (k >= 31);
    const int tt = t + (k - (c ? 31 : 0)) - 15;
    float v = 0.f;
    if (tt >= 0 && tt < 512) v = (c ? attc : att)[(b << 9) + tt];
    return v;
}

// ---------------------------------------------------------------------------
// kernel 1: p_annot[m][a] = inputs[m][:] . annot_w[a][:] + annot_b[a], m = b*T+t
// ---------------------------------------------------------------------------
__global__ __launch_bounds__(256) void pannot_kernel(const float* __restrict__ inputs,
                                                     const float* __restrict__ annot_w,
                                                     const float* __restrict__ annot_b,
                                                     float* __restrict__ pannot)
{
    const int wave = threadIdx.x >> 5, lane = threadIdx.x & 31;
    const int tile = blockIdx.x * 8 + wave;             // 1024 blocks * 8 waves = 8192 tiles
    const int m0 = (tile >> 3) << 4;
    const int n0 = (tile & 7) << 4;
    const int row = lane & 15, half = lane >> 4;
    v8f c = {0.f, 0.f, 0.f, 0.f, 0.f, 0.f, 0.f, 0.f};
    const float* ap = inputs + (size_t)(m0 + row) * IN_ + 2 * half;
    const float* wp = annot_w + (size_t)(n0 + row) * IN_ + 2 * half;
    for (int k = 0; k < IN_; k += 4) {
        v2f a = *(const v2f*)(ap + k);
        v2f b = *(const v2f*)(wp + k);
        c = __builtin_amdgcn_wmma_f32_16x16x4_f32(false, a, false, b,
                                                  (short)0, c, false, false);
    }
    const int col = n0 + row;
    const float bv = annot_b[col];
#pragma unroll
    for (int i = 0; i < 8; ++i)
        pannot[(size_t)(m0 + i + 8 * half) * A_ + col] = c[i] + bv;
}

// kernel 2: Weff[a][kk] = sum_f loc_lin_w[a][f] * conv_w[f][c][k31], padded to K=64
__global__ __launch_bounds__(256) void weff_kernel(const float* __restrict__ conv_w,
                                                   const float* __restrict__ lin_w,
                                                   float* __restrict__ weff)
{
    const int idx = blockIdx.x * 256 + threadIdx.x;     // 32 blocks -> 8192
    const int a = idx >> 6, kk = idx & 63;
    float acc = 0.f;
    if (kk < 62) {
        const int c = (kk >= 31);
        const int k31 = kk - (c ? 31 : 0);
        for (int f = 0; f < 32; ++f)
            acc += lin_w[a * 32 + f] * conv_w[(f * 2 + c) * 31 + k31];
    }
    weff[idx] = acc;
}

// ---------------------------------------------------------------------------
// kernel 3: persistent decoder — 1 workgroup, 1024 threads (32 wave32 waves)
// ---------------------------------------------------------------------------
__global__ __launch_bounds__(1024, 1) void decoder_kernel(
    const float* __restrict__ inputs, const float* __restrict__ memory,
    const float* __restrict__ prenet_w1, const float* __restrict__ prenet_b1,
    const float* __restrict__ prenet_w2, const float* __restrict__ prenet_b2,
    const float* __restrict__ arnn_wih, const float* __restrict__ arnn_whh,
    const float* __restrict__ arnn_bih, const float* __restrict__ arnn_bhh,
    const float* __restrict__ loc_lin_b,
    const float* __restrict__ query_w, const float* __restrict__ query_b,
    const float* __restrict__ v_w,
    const float* __restrict__ proj_w, const float* __restrict__ proj_b,
    const float* __restrict__ d1_wih, const float* __restrict__ d1_whh,
    const float* __restrict__ d1_bih, const float* __restrict__ d1_bhh,
    const float* __restrict__ d2_wih, const float* __restrict__ d2_whh,
    const float* __restrict__ d2_bih, const float* __restrict__ d2_bhh,
    const float* __restrict__ mel_w, const float* __restrict__ mel_b,
    const float* __restrict__ arnn_init, const float* __restrict__ mem_init,
    const float* __restrict__ dec_init, const float* __restrict__ stop_w,
    const float* __restrict__ stop_b,
    float* __restrict__ ws, float* __restrict__ out)
{
    extern __shared__ float lds[];
    const int tid = threadIdx.x;
    const int wave = tid >> 5, lane = tid & 31;
    const int row = lane & 15, half = lane >> 4;

    // ---- init states / carries --------------------------------------------
    for (int idx = tid; idx < B_ * T_; idx += 1024) { ws[WS_ATT + idx] = 0.f; ws[WS_ATTC + idx] = 0.f; }
    for (int idx = tid; idx < 32 * 256; idx += 1024) {
        int b = idx >> 8, j = idx & 255;
        lds[L_HA  + b * 258 + j] = arnn_init[j];
        lds[L_HD1 + b * 258 + j] = dec_init[j];
        lds[L_HD2 + b * 258 + j] = dec_init[256 + j];
        lds[L_CTX + b * 258 + j] = 0.f;
    }
    __threadfence(); __syncthreads(); __threadfence();

    for (int step = 0; step < TDEC; ++step) {
        // ===== prenet: stage mem_in (562-pitch) at L_GI =====================
        for (int idx = tid; idx < 32 * KPRE; idx += 1024) {
            int b = idx / KPRE, k = idx - b * KPRE;
            float v = (step == 0) ? mem_init[k]
                                  : memory[(size_t)b * 56000 + (size_t)(step - 1) * 560 + k];
            lds[L_GI + b * 562 + k] = v;
        }
        __syncthreads();
        gemm_lds(lds, L_GI, 562, prenet_w1, prenet_b1, lds, L_GHB, 258, 256, 560, 1, wave, lane);
        __syncthreads();
        gemm_lds(lds, L_GHB, 258, prenet_w2, prenet_b2, lds, L_XBUF, 386, 128, 256, 1, wave, lane);
        for (int idx = tid; idx < 32 * 256; idx += 1024) {   // xbuf[:,128:384] = ctx
            int b = idx >> 8, j = idx & 255;
            lds[L_XBUF + b * 386 + 128 + j] = lds[L_CTX + b * 258 + j];
        }
        __syncthreads();

        // ===== attention GRU ===============================================
        gru(lds, L_HA, L_XBUF, 386, 384, arnn_wih, arnn_whh, arnn_bih, arnn_bhh, wave, lane, tid);

        // ===== attention: p_q, folded-conv GEMM, energies, context =========
        gemm_lds(lds, L_HA, 258, query_w, query_b, lds, L_GI, 130, 128, 256, 0, wave, lane);
        if (tid < 128) lds[L_GI + 32 * 130 + tid] = v_w[tid];     // v cache
        if (tid < 64)  lds[L_RED + tid] = 0.f;
        __syncthreads();

        {   // p_loc = im2col([attn,attn_cum]) @ Weff^T + loc_lin_b  -> ws.ploc
            const float* att  = ws + WS_ATT;
            const float* attc = ws + WS_ATTC;
            const float* weff = ws + WS_WEFF;
            float* ploc = ws + WS_PLOC;
            for (int tile = wave; tile < 8192; tile += 32) {
                const int m0 = (tile & 1023) << 4;
                const int n0 = (tile >> 10) << 4;
                const int m = m0 + row, b = m >> 9, t = m & 511;
                v8f c = {0.f, 0.f, 0.f, 0.f, 0.f, 0.f, 0.f, 0.f};
                const float* wp = weff + (n0 + row) * 64 + 2 * half;
                for (int k0 = 0; k0 < 64; k0 += 4) {
                    const int kk = k0 + 2 * half;
                    v2f a; a.x = im2col(att, attc, b, t, kk);
                           a.y = im2col(att, attc, b, t, kk + 1);
                    v2f bb = *(const v2f*)(wp + k0);
                    c = __builtin_amdgcn_wmma_f32_16x16x4_f32(false, a, false, bb,
                                                              (short)0, c, false, false);
                }
                const int col = n0 + row;
                const float bv = loc_lin_b[col];
#pragma unroll
                for (int i = 0; i < 8; ++i)
                    ploc[(size_t)(m0 + i + 8 * half) * A_ + col] = c[i] + bv;
            }
        }
        __threadfence(); __syncthreads(); __threadfence();

        // energies: e = v . tanh(pq + p_annot + p_loc); sig + row sums
        for (int i = 0; i < 16; ++i) {
            const int pos = tid + (i << 10);
            const int b = pos >> 9;
            const float* pa = ws + WS_PANNOT + (size_t)pos * A_;
            const float* pl = ws + WS_PLOC   + (size_t)pos * A_;
            float acc = 0.f;
            for (int a = 0; a < A_; ++a)
                acc += lds[L_GI + 32 * 130 + a] *
                       tanhf(lds[L_GI + b * 130 + a] + pa[a] + pl[a]);
            const float s = 1.f / (1.f + expf(-acc));
            ws[WS_ATT + pos] = s;
            atomicAdd(&lds[L_RED + b], s);
        }
        __syncthreads();
        for (int i = 0; i < 16; ++i) {                  // normalize, cum, output
            const int pos = tid + (i << 10);
            const int b = pos >> 9, t = pos & 511;
            const float a = ws[WS_ATT + pos] / lds[L_RED + b];
            ws[WS_ATT + pos] = a;
            ws[WS_ATTC + pos] += a;
            out[OUT_ATTNS + ((size_t)b * TDEC + step) * T_ + t] = a;
        }
        __threadfence(); __syncthreads(); __threadfence();
        {   // context: ctx[b][d] = sum_t a[b][t] * inputs[b][t][d]
            const int b = tid >> 5;
            const int dg = (tid & 31) << 3;
            float acc[8] = {0.f, 0.f, 0.f, 0.f, 0.f, 0.f, 0.f, 0.f};
            const float* ar = ws + WS_ATT + b * T_;
            const float* ip = inputs + (size_t)b * T_ * IN_ + dg;
            for (int t = 0; t < T_; ++t) {
                const float av = ar[t];
                const float* p = ip + (size_t)t * IN_;
#pragma unroll
                for (int d = 0; d < 8; ++d) acc[d] += av * p[d];
            }
#pragma unroll
            for (int d = 0; d < 8; ++d) lds[L_CTX + b * 258 + dg + d] = acc[d];
        }
        __syncthreads();

        // ===== decoder =====================================================
        for (int idx = tid; idx < 32 * 512; idx += 1024) {   // cbuf = [h_a | ctx]
            int b = idx >> 9, j = idx & 511;
            lds[L_GI + b * 514 + j] = (j < 256) ? lds[L_HA + b * 258 + j]
                                                : lds[L_CTX + b * 258 + (j - 256)];
        }
        __syncthreads();
        gemm_lds(lds, L_GI, 514, proj_w, proj_b, lds, L_XBUF, 258, 256, 512, 0, wave, lane);
        __syncthreads();
        gru(lds, L_HD1, L_XBUF, 258, 256, d1_wih, d1_whh, d1_bih, d1_bhh, wave, lane, tid);
        for (int idx = tid; idx < 32 * 256; idx += 1024) {   // dec_in += d1_h
            int b = idx >> 8, j = idx & 255;
            lds[L_XBUF + b * 258 + j] += lds[L_HD1 + b * 258 + j];
        }
        __syncthreads();
        gru(lds, L_HD2, L_XBUF, 258, 256, d2_wih, d2_whh, d2_bih, d2_bhh, wave, lane, tid);
        for (int idx = tid; idx < 32 * 256; idx += 1024) {   // dec_out at L_GHB
            int b = idx >> 8, j = idx & 255;
            lds[L_GHB + b * 258 + j] = lds[L_HD2 + b * 258 + j] + lds[L_XBUF + b * 258 + j];
        }
        __syncthreads();
        // mel: sigmoid(dec_out @ mel_w^T + mel_b) -> L_GI [32][562]
        gemm_lds(lds, L_GHB, 258, mel_w, mel_b, lds, L_GI, 562, 560, 256, 2, wave, lane);
        __syncthreads();
        for (int idx = tid; idx < 32 * MELN; idx += 1024) {
            int b = idx / MELN, j = idx - b * MELN;
            out[OUT_OUTS + ((size_t)b * TDEC + step) * MELN + j] = lds[L_GI + b * 562 + j];
        }
        if (tid < 32) {                                     // stopnet (N=1, VALU dot)
            const int b = tid;
            float acc = stop_b[0];
            for (int c = 0; c < 256; ++c)  acc += stop_w[c] * lds[L_GHB + b * 258 + c];
            for (int j = 0; j < MELN; ++j) acc += stop_w[256 + j] * lds[L_GI + b * 562 + j];
            out[OUT_STOPS + (size_t)b * TDEC + step] = 1.f / (1.f + expf(-acc));
        }
        __syncthreads();   // protect L_GI/L_GHB reuse next step
    }
}

// ---------------------------------------------------------------------------
extern "C" void kernel_launch(void* const* d_in, const int* in_sizes, int n_in,
                              void* d_out, int out_size, void* d_ws, size_t ws_size,
                              hipStream_t stream)
{
    const float* inputs    = (const float*)d_in[0];
    const float* memory    = (const float*)d_in[1];
    const float* prenet_w1 = (const float*)d_in[2];
    const float* prenet_b1 = (const float*)d_in[3];
    const float* prenet_w2 = (const float*)d_in[4];
    const float* prenet_b2 = (const float*)d_in[5];
    const float* arnn_wih  = (const float*)d_in[6];
    const float* arnn_whh  = (const float*)d_in[7];
    const float* arnn_bih  = (const float*)d_in[8];
    const float* arnn_bhh  = (const float*)d_in[9];
    const float* loc_conv_w= (const float*)d_in[10];
    const float* loc_lin_w = (const float*)d_in[11];
    const float* loc_lin_b = (const float*)d_in[12];
    const float* query_w   = (const float*)d_in[13];
    const float* query_b   = (const float*)d_in[14];
    const float* annot_w   = (const float*)d_in[15];
    const float* annot_b   = (const float*)d_in[16];
    const float* v_w       = (const float*)d_in[17];
    const float* proj_w    = (const float*)d_in[18];
    const float* proj_b    = (const float*)d_in[19];
    const float* d1_wih    = (const float*)d_in[20];
    const float* d1_whh    = (const float*)d_in[21];
    const float* d1_bih    = (const float*)d_in[22];
    const float* d1_bhh    = (const float*)d_in[23];
    const float* d2_wih    = (const float*)d_in[24];
    const float* d2_whh    = (const float*)d_in[25];
    const float* d2_bih    = (const float*)d_in[26];
    const float* d2_bhh    = (const float*)d_in[27];
    const float* mel_w     = (const float*)d_in[28];
    const float* mel_b     = (const float*)d_in[29];
    const float* arnn_init = (const float*)d_in[30];
    const float* mem_init  = (const float*)d_in[31];
    const float* dec_init  = (const float*)d_in[32];
    const float* stop_w    = (const float*)d_in[33];
    const float* stop_b    = (const float*)d_in[34];

    float* ws  = (float*)d_ws;
    float* out = (float*)d_out;

    pannot_kernel<<<1024, 256, 0, stream>>>(inputs, annot_w, annot_b, ws + WS_PANNOT);
    weff_kernel<<<32, 256, 0, stream>>>(loc_conv_w, loc_lin_w, ws + WS_WEFF);

    const size_t ldsBytes = (size_t)L_TOT * sizeof(float);   // 313,344 B < 320KB WGP LDS
    decoder_kernel<<<1, 1024, ldsBytes, stream>>>(
        inputs, memory, prenet_w1, prenet_b1, prenet_w2, prenet_b2,
        arnn_wih, arnn_whh, arnn_bih, arnn_bhh, loc_lin_b,
        query_w, query_b, v_w, proj_w, proj_b,
        d1_wih, d1_whh, d1_bih, d1_bhh, d2_wih, d2_whh, d2_bih, d2_bhh,
        mel_w, mel_b, arnn_init, mem_init, dec_init, stop_w, stop_b,
        ws, out);
}